// SparseLinearEnhanced_23716809408770
// MI455X (gfx1250) — compile-verified
//
#include <hip/hip_runtime.h>

// ---------------- problem constants (match reference) ----------------
#define IN_F   4096
#define OUT_F  4096
#define BATCH  4096

typedef float v2f __attribute__((ext_vector_type(2)));
typedef float v8f __attribute__((ext_vector_type(8)));
typedef int   v4i __attribute__((ext_vector_type(4)));

constexpr int BM = 128;          // batch rows per block
constexpr int BN = 128;          // out features per block
constexpr int BK = 32;           // K chunk per LDS stage
constexpr int LDSS = BK + 4;     // padded LDS row stride (36 floats, keeps 16B align)

// ---- CDNA5 async global->LDS copy (ASYNCcnt path), with safe fallback ----
#if __has_builtin(__builtin_amdgcn_global_load_async_to_lds_b128) && \
    __has_builtin(__builtin_amdgcn_s_wait_asynccnt)
#define USE_ASYNC_COPY 1
#else
#define USE_ASYNC_COPY 0
#endif

__device__ __forceinline__ void copy16_to_lds(float* lds_dst, const float* gsrc) {
#if USE_ASYNC_COPY
    // GLOBAL_LOAD_ASYNC_TO_LDS_B128: per-lane 16B memory->LDS, no VGPR data path.
    // Builtin signature (from clang diagnostic): (v4i AS1*, v4i AS3*, imm, imm)
    __builtin_amdgcn_global_load_async_to_lds_b128(
        (__attribute__((address_space(1))) v4i*)(void*)gsrc,
        (__attribute__((address_space(3))) v4i*)(void*)lds_dst,
        /*imm offset=*/0, /*cpol=*/0);
#else
    *(float4*)lds_dst = *(const float4*)gsrc;
#endif
}

__device__ __forceinline__ void wait_async_copies() {
#if USE_ASYNC_COPY
    __builtin_amdgcn_s_wait_asynccnt(0);   // S_WAIT_ASYNCCNT 0
#endif
}

// ---------------- phase 1: zero the dense W workspace ----------------
__global__ __launch_bounds__(256) void zero_ws(float* __restrict__ W) {
    size_t i = ((size_t)blockIdx.x * 256 + threadIdx.x) * 4;
    float4 z = make_float4(0.f, 0.f, 0.f, 0.f);
    *(float4*)(W + i) = z;
}

// ---------------- phase 2: COO scatter-add into dense W --------------
__global__ __launch_bounds__(256) void scatter_coo(float* __restrict__ W,
                                                   const float* __restrict__ w,
                                                   const int* __restrict__ conn,
                                                   int nnz) {
    int e = blockIdx.x * 256 + threadIdx.x;
    if (e >= nnz) return;
    int o = conn[e];          // connectivity row 0: out index
    int i = conn[nnz + e];    // connectivity row 1: in index
    atomicAdd(&W[(size_t)o * IN_F + i], w[e]);   // GLOBAL_ATOMIC_ADD_F32
}

// ---------------- phase 3: fp32 WMMA GEMM  y = x @ W^T + bias --------
__global__ __launch_bounds__(256) void spmm_wmma(float* __restrict__ y,
                                                 const float* __restrict__ x,
                                                 const float* __restrict__ W,
                                                 const float* __restrict__ bias) {
    __shared__ float As[2][BM][LDSS];   // double-buffered x tile [m][k]
    __shared__ float Bs[2][BN][LDSS];   // double-buffered W tile [n][k]

    const int t    = threadIdx.x;
    const int lane = t & 31;
    const int wv   = t >> 5;       // wave 0..7
    const int wm   = wv >> 2;      // 0..1  -> 64-row slab
    const int wn   = wv & 3;       // 0..3  -> 32-col slab

    const int blockN = blockIdx.x * BN;
    const int blockM = blockIdx.y * BM;

    // staging map: 256 threads * 4 * 16B = one 128x32 fp32 tile
    const int lrow = t >> 3;           // 0..31
    const int lcol = (t & 7) * 4;      // 0,4,...,28

    // WMMA 16x16x4 f32 fragment coordinates
    const int frag_m = lane & 15;
    const int frag_k = (lane >> 4) * 2;

    const float* xg = x + (size_t)blockM * IN_F;
    const float* Wg = W + (size_t)blockN * IN_F;

    v8f acc[4][2] = {};

    // prologue: stage K-chunk 0 into buffer 0
    #pragma unroll
    for (int r = 0; r < 4; ++r) {
        int row = lrow + r * 32;
        copy16_to_lds(&As[0][row][lcol], xg + (size_t)row * IN_F + lcol);
        copy16_to_lds(&Bs[0][row][lcol], Wg + (size_t)row * IN_F + lcol);
    }

    for (int k0 = 0; k0 < IN_F; k0 += BK) {
        const int cur = (k0 / BK) & 1;

        // my async copies into `cur` done; barrier makes everyone's visible
        wait_async_copies();
        __syncthreads();

        // kick off next chunk into the other buffer while we compute
        if (k0 + BK < IN_F) {
            const int nxt = cur ^ 1;
            #pragma unroll
            for (int r = 0; r < 4; ++r) {
                int row = lrow + r * 32;
                copy16_to_lds(&As[nxt][row][lcol],
                              xg + (size_t)row * IN_F + (k0 + BK) + lcol);
                copy16_to_lds(&Bs[nxt][row][lcol],
                              Wg + (size_t)row * IN_F + (k0 + BK) + lcol);
            }
        }

        // 8 k-steps of V_WMMA_F32_16X16X4_F32 on buffer `cur`
        #pragma unroll
        for (int ks = 0; ks < BK; ks += 4) {
            v2f a[4], b[2];
            #pragma unroll
            for (int mt = 0; mt < 4; ++mt)
                a[mt] = *(const v2f*)&As[cur][wm * 64 + mt * 16 + frag_m][ks + frag_k];
            #pragma unroll
            for (int nt = 0; nt < 2; ++nt)
                b[nt] = *(const v2f*)&Bs[cur][wn * 32 + nt * 16 + frag_m][ks + frag_k];
            #pragma unroll
            for (int mt = 0; mt < 4; ++mt)
                #pragma unroll
                for (int nt = 0; nt < 2; ++nt)
                    acc[mt][nt] = __builtin_amdgcn_wmma_f32_16x16x4_f32(
                        false, a[mt], false, b[nt],
                        (short)0, acc[mt][nt], false, false);
        }
        // no trailing barrier: next iteration's wait+barrier orders buffer reuse
    }

    // ---- epilogue: add bias, store C (16x16 f32 D layout) ----
    const int n_lane = lane & 15;
    const int m_off  = (lane >> 4) * 8;
    #pragma unroll
    for (int nt = 0; nt < 2; ++nt) {
        int col  = blockN + wn * 32 + nt * 16 + n_lane;
        float bv = bias[col];
        #pragma unroll
        for (int mt = 0; mt < 4; ++mt) {
            int row0 = blockM + wm * 64 + mt * 16 + m_off;
            #pragma unroll
            for (int r = 0; r < 8; ++r)
                y[(size_t)(row0 + r) * OUT_F + col] = acc[mt][nt][r] + bv;
        }
    }
}

// ---------------- launcher ----------------
extern "C" void kernel_launch(void* const* d_in, const int* in_sizes, int n_in,
                              void* d_out, int out_size, void* d_ws, size_t ws_size,
                              hipStream_t stream) {
    const float* x    = (const float*)d_in[0];
    const float* w    = (const float*)d_in[1];
    const float* bias = (const float*)d_in[2];
    const int*   conn = (const int*)d_in[3];
    float* y = (float*)d_out;
    float* W = (float*)d_ws;              // dense 4096x4096 fp32 = 64 MB scratch

    const int nnz = in_sizes[1];

    zero_ws<<<(OUT_F * IN_F) / (256 * 4), 256, 0, stream>>>(W);
    scatter_coo<<<(nnz + 255) / 256, 256, 0, stream>>>(W, w, conn, nnz);
    spmm_wmma<<<dim3(OUT_F / BN, BATCH / BM), 256, 0, stream>>>(y, x, W, bias);
}